// QKVGATLayer_46437186404440
// MI455X (gfx1250) — compile-verified
//
#include <hip/hip_runtime.h>
#include <math.h>

typedef __attribute__((ext_vector_type(2))) float v2f;
typedef __attribute__((ext_vector_type(8))) float v8f;

#define Fdim 128
#define Hn 4
#define Dh 32
#define SCALE 0.17677669529663687f   // 1/sqrt(32)
#define LN_EPS 1e-5f

// ---------------------------------------------------------------------------
// init: mx = -inf, sm = 0, agg = 0
// ---------------------------------------------------------------------------
__global__ void init_ws_kernel(float* __restrict__ mx, float* __restrict__ sm,
                               float* __restrict__ agg, int nmx, int nagg) {
    int idx = blockIdx.x * blockDim.x + threadIdx.x;
    if (idx < nmx) {
        mx[idx] = -INFINITY;
        sm[idx] = 0.0f;
    }
    if (idx < nagg) agg[idx] = 0.0f;
}

// ---------------------------------------------------------------------------
// Fused projection GEMM via V_WMMA_F32_16X16X4_F32 (exact fp32).
// Output layout: [Q | K | V | R] each N x 128, row-major. R includes b_res.
// One wave computes one 16x16 tile of one of the four products.
// ---------------------------------------------------------------------------
__global__ void proj_gemm_kernel(const float* __restrict__ X,
                                 const float* __restrict__ WQ,
                                 const float* __restrict__ WK,
                                 const float* __restrict__ WV,
                                 const float* __restrict__ WR,
                                 const float* __restrict__ bres,
                                 float* __restrict__ out,   // 4*N*128 floats
                                 int N, int rowTiles) {
    const int lane = threadIdx.x & 31;
    const int waveInBlock = threadIdx.x >> 5;
    const int wavesPerBlock = blockDim.x >> 5;
    const int tile = blockIdx.x * wavesPerBlock + waveInBlock;
    const int totalTiles = rowTiles * 32;       // 32 col-tiles = 4 mats * 8
    if (tile >= totalTiles) return;             // wave-uniform guard

    const int rt = tile >> 5;                   // row tile
    const int ct = tile & 31;                   // col tile across 4 matrices
    const int mat = ct >> 3;                    // 0=Q 1=K 2=V 3=R
    const int col = (ct & 7) << 4;              // column base within matrix

    const float* __restrict__ W =
        (mat == 0) ? WQ : (mat == 1) ? WK : (mat == 2) ? WV : WR;

    const int m = lane & 15;                    // row-in-tile for A, col for B/C
    const int g = lane >> 4;                    // half-wave group
    const int row0 = rt << 4;
    int arow = row0 + m;
    if (arow >= N) arow = N - 1;                // clamp loads; EXEC stays all-1s

    const float* __restrict__ xrow = X + (size_t)arow * Fdim;
    const float* __restrict__ wcol = W + col + m;

    v8f acc = {};
#pragma unroll
    for (int k0 = 0; k0 < Fdim; k0 += 4) {
        // A fragment: a[v] = X[m][k0 + v + 2g]  -> contiguous pair
        v2f a;
        a.x = xrow[k0 + 2 * g];
        a.y = xrow[k0 + 2 * g + 1];
        // B fragment: b[v] = W[k0 + v + 2g][col + m]
        v2f b;
        b.x = wcol[(size_t)(k0 + 2 * g) * Fdim];
        b.y = wcol[(size_t)(k0 + 2 * g + 1) * Fdim];
        acc = __builtin_amdgcn_wmma_f32_16x16x4_f32(
            /*neg_a=*/false, a, /*neg_b=*/false, b,
            /*c_mod=*/(short)0, acc, /*reuse_a=*/false, /*reuse_b=*/false);
    }

    if (mat == 3) {
        const float bv = bres[col + m];
#pragma unroll
        for (int v = 0; v < 8; ++v) acc[v] += bv;
    }

    float* __restrict__ obase = out + (size_t)mat * N * Fdim + col + m;
#pragma unroll
    for (int v = 0; v < 8; ++v) {
        const int r = row0 + v + 8 * g;          // C/D layout: row = v + 8g
        if (r < N) obase[(size_t)r * Fdim] = acc[v];
    }
}

// ---------------------------------------------------------------------------
// Float atomic max via signed/unsigned integer atomics (works with -inf init)
// ---------------------------------------------------------------------------
__device__ __forceinline__ void atomicMaxF(float* addr, float val) {
    if (val >= 0.0f)
        atomicMax((int*)addr, __float_as_int(val));
    else
        atomicMin((unsigned int*)addr, (unsigned int)__float_as_int(val));
}

// ---------------------------------------------------------------------------
// Pass A: per (edge, head) attention score + segment max over dst
// ---------------------------------------------------------------------------
__global__ void edge_score_kernel(const float* __restrict__ Q,
                                  const float* __restrict__ K,
                                  const int* __restrict__ src,
                                  const int* __restrict__ dst,
                                  float* __restrict__ score,   // E*4
                                  float* __restrict__ mx,      // N*4
                                  int E) {
    const int idx = blockIdx.x * blockDim.x + threadIdx.x;
    if (idx >= E * Hn) return;
    const int e = idx >> 2;
    const int h = idx & 3;
    const int s = src[e];
    const int d = dst[e];
    const float4* __restrict__ q4 = (const float4*)(Q + (size_t)d * Fdim + h * Dh);
    const float4* __restrict__ k4 = (const float4*)(K + (size_t)s * Fdim + h * Dh);
    float acc = 0.0f;
#pragma unroll
    for (int i = 0; i < Dh / 4; ++i) {
        const float4 q = q4[i];
        const float4 k = k4[i];
        acc += q.x * k.x + q.y * k.y + q.z * k.z + q.w * k.w;
    }
    acc *= SCALE;
    score[idx] = acc;
    atomicMaxF(&mx[d * Hn + h], acc);
}

// ---------------------------------------------------------------------------
// Pass B: ex = exp(score - mx[dst]); segment sum over dst; score := ex
// ---------------------------------------------------------------------------
__global__ void edge_exp_kernel(const int* __restrict__ dst,
                                const float* __restrict__ mx,
                                float* __restrict__ score,    // in: score, out: ex
                                float* __restrict__ sm,       // N*4
                                int E) {
    const int idx = blockIdx.x * blockDim.x + threadIdx.x;
    if (idx >= E * Hn) return;
    const int e = idx >> 2;
    const int h = idx & 3;
    const int d = dst[e];
    const float ex = expf(score[idx] - mx[d * Hn + h]);
    score[idx] = ex;
    atomicAdd(&sm[d * Hn + h], ex);
}

// ---------------------------------------------------------------------------
// Pass C: agg[dst, h, :] += alpha * V[src, h, :]   (one thread per (e, feature))
// ---------------------------------------------------------------------------
__global__ void edge_agg_kernel(const int* __restrict__ src,
                                const int* __restrict__ dst,
                                const float* __restrict__ Vm,
                                const float* __restrict__ ex,  // E*4
                                const float* __restrict__ sm,  // N*4
                                float* __restrict__ agg,       // N*128
                                int E) {
    const long long idx = (long long)blockIdx.x * blockDim.x + threadIdx.x;
    if (idx >= (long long)E * Fdim) return;
    const int e = (int)(idx >> 7);
    const int hd = (int)(idx & 127);
    const int h = hd >> 5;
    const int s = src[e];
    const int d = dst[e];
    const float alpha = ex[e * Hn + h] / (sm[d * Hn + h] + 1e-12f);
    const float val = alpha * Vm[(size_t)s * Fdim + hd];
    atomicAdd(&agg[(size_t)d * Fdim + hd], val);
}

// ---------------------------------------------------------------------------
// Finalize: H = agg + R (R already has nodes@W_res + b_res), then LayerNorm
// One 128-thread block per node.
// ---------------------------------------------------------------------------
__global__ void finalize_ln_kernel(const float* __restrict__ agg,
                                   const float* __restrict__ R,
                                   const float* __restrict__ gamma,
                                   const float* __restrict__ beta,
                                   float* __restrict__ out, int N) {
    const int nid = blockIdx.x;
    const int j = threadIdx.x;
    if (nid >= N) return;
    __shared__ float red[Fdim];

    const float h = agg[(size_t)nid * Fdim + j] + R[(size_t)nid * Fdim + j];

    red[j] = h;
    __syncthreads();
#pragma unroll
    for (int s = Fdim / 2; s > 0; s >>= 1) {
        if (j < s) red[j] += red[j + s];
        __syncthreads();
    }
    const float mu = red[0] * (1.0f / Fdim);
    __syncthreads();

    const float dv = h - mu;
    red[j] = dv * dv;
    __syncthreads();
#pragma unroll
    for (int s = Fdim / 2; s > 0; s >>= 1) {
        if (j < s) red[j] += red[j + s];
        __syncthreads();
    }
    const float var = red[0] * (1.0f / Fdim);

    const float xh = dv * rsqrtf(var + LN_EPS);
    out[(size_t)nid * Fdim + j] = gamma[j] * xh + beta[j];
}

// ---------------------------------------------------------------------------
extern "C" void kernel_launch(void* const* d_in, const int* in_sizes, int n_in,
                              void* d_out, int out_size, void* d_ws, size_t ws_size,
                              hipStream_t stream) {
    const float* nodes = (const float*)d_in[0];
    const float* WQ    = (const float*)d_in[1];
    const float* WK    = (const float*)d_in[2];
    const float* WV    = (const float*)d_in[3];
    const float* WR    = (const float*)d_in[4];
    const float* bres  = (const float*)d_in[5];
    const float* gamma = (const float*)d_in[6];
    const float* beta  = (const float*)d_in[7];
    const int*   ei    = (const int*)d_in[8];

    const int N = in_sizes[0] / Fdim;
    const int E = in_sizes[8] / 2;
    const int* src = ei;          // edge_index[0]
    const int* dst = ei + E;      // edge_index[1]

    // workspace layout (floats)
    float* ws   = (float*)d_ws;
    float* QKVR = ws;                                   // 4*N*128
    float* sc   = QKVR + (size_t)4 * N * Fdim;          // E*4
    float* mx   = sc + (size_t)E * Hn;                  // N*4
    float* sm   = mx + (size_t)N * Hn;                  // N*4
    float* agg  = sm + (size_t)N * Hn;                  // N*128
    const float* Qm = QKVR;
    const float* Km = QKVR + (size_t)N * Fdim;
    const float* Vm = QKVR + (size_t)2 * N * Fdim;
    const float* Rm = QKVR + (size_t)3 * N * Fdim;

    // init mx/sm/agg
    {
        const int nmx = N * Hn;
        const int nagg = N * Fdim;
        const int nmax = (nagg > nmx) ? nagg : nmx;
        init_ws_kernel<<<(nmax + 255) / 256, 256, 0, stream>>>(mx, sm, agg, nmx, nagg);
    }

    // fused WMMA projection GEMM: [Q|K|V|R] = X @ [WQ|WK|WV|WR] (+ b_res on R)
    {
        const int rowTiles = (N + 15) / 16;
        const int tiles = rowTiles * 32;
        const int wavesPerBlock = 8;                    // 256 threads
        const int blocks = (tiles + wavesPerBlock - 1) / wavesPerBlock;
        proj_gemm_kernel<<<blocks, 256, 0, stream>>>(nodes, WQ, WK, WV, WR, bres,
                                                     QKVR, N, rowTiles);
    }

    // edge phase
    {
        const long long eh = (long long)E * Hn;
        const int blk = 256;
        edge_score_kernel<<<(int)((eh + blk - 1) / blk), blk, 0, stream>>>(
            Qm, Km, src, dst, sc, mx, E);
        edge_exp_kernel<<<(int)((eh + blk - 1) / blk), blk, 0, stream>>>(
            dst, mx, sc, sm, E);
        const long long tot = (long long)E * Fdim;
        edge_agg_kernel<<<(int)((tot + blk - 1) / blk), blk, 0, stream>>>(
            src, dst, Vm, sc, sm, agg, E);
    }

    // residual + LayerNorm
    finalize_ln_kernel<<<N, Fdim, 0, stream>>>(agg, Rm, gamma, beta, (float*)d_out, N);
}